// Attention_56650618635094
// MI455X (gfx1250) — compile-verified
//
#include <hip/hip_runtime.h>
#include <math.h>
#include <stdint.h>

// ---------------------------------------------------------------------------
// Shapes: x (64,197,32,32) f32; Wq/Wk/Wv (197,1,11,11) f32; out (64,197,32,32)
// Pipeline: dwconv(q,k,vT) -> alpha = q.kT/32 (WMMA f32) -> row softmax
//           -> out = alpha.v (WMMA f32, vT stored so both GEMMs are A.B^T)
// GEMM: 8 waves/block, wave-level 32x32 register blocking (2x2 WMMA tiles),
//       double-buffered LDS tiles filled with GLOBAL_LOAD_ASYNC_TO_LDS_B128
//       (ASYNCcnt-tracked) so the next K-chunk streams in during the WMMAs.
// ---------------------------------------------------------------------------

typedef __attribute__((ext_vector_type(2))) float v2f;
typedef __attribute__((ext_vector_type(8))) float v8f;
typedef __attribute__((ext_vector_type(4))) int   v4i;

#define NB   64
#define NT   197
#define HW   1024
#define TP   200          // padded trailing dim for vT / alpha (16B-aligned rows)

// ---------------- CDNA5 async global->LDS copy helpers ----------------------
__device__ __forceinline__ void async_copy_b128(const float* g, float* l)
{
#if defined(__has_builtin) && __has_builtin(__builtin_amdgcn_global_load_async_to_lds_b128)
    __builtin_amdgcn_global_load_async_to_lds_b128(
        (__attribute__((address_space(1))) v4i*)(uintptr_t)g,
        (__attribute__((address_space(3))) v4i*)(uintptr_t)l,
        0, 0);
#else
    unsigned loff = (unsigned)(uintptr_t)l;              // low 32 bits = LDS offset
    unsigned long long ga = (unsigned long long)(uintptr_t)g;
    asm volatile("global_load_async_to_lds_b128 %0, %1, off"
                 :: "v"(loff), "v"(ga) : "memory");
#endif
}

__device__ __forceinline__ void async_wait0()
{
#if defined(__has_builtin) && __has_builtin(__builtin_amdgcn_s_wait_asynccnt)
    __builtin_amdgcn_s_wait_asynccnt(0);
#else
    asm volatile("s_wait_asynccnt 0" ::: "memory");
#endif
}

// ---------------- Kernel 1: depthwise 11x11 conv -> q, k, vT ----------------
__global__ __launch_bounds__(256)
void dwconv_qkv(const float* __restrict__ x,
                const float* __restrict__ Wq,
                const float* __restrict__ Wk,
                const float* __restrict__ Wv,
                float* __restrict__ q, float* __restrict__ k,
                float* __restrict__ vT)
{
    const int c = blockIdx.x;          // token/channel 0..196
    const int b = blockIdx.y;          // batch 0..63
    const int tid = threadIdx.x;

    __shared__ __align__(16) float xs[42 * 42];   // 32 + 2*5 halo
    __shared__ float wq[121], wk[121], wv[121];

    const float* xp = x + ((long)b * NT + c) * HW;

    for (int i = tid; i < 42 * 42; i += 256) {
        int hy = i / 42 - 5;
        int wx = i % 42 - 5;
        xs[i] = (hy >= 0 && hy < 32 && wx >= 0 && wx < 32) ? xp[hy * 32 + wx] : 0.0f;
    }
    if (tid < 121) {
        wq[tid] = Wq[c * 121 + tid];
        wk[tid] = Wk[c * 121 + tid];
        wv[tid] = Wv[c * 121 + tid];
    }
    __syncthreads();

    for (int p = tid; p < HW; p += 256) {
        int h = p >> 5, w = p & 31;
        float aq = 0.f, ak = 0.f, av = 0.f;
#pragma unroll
        for (int dy = 0; dy < 11; ++dy) {
#pragma unroll
            for (int dx = 0; dx < 11; ++dx) {
                float xv = xs[(h + dy) * 42 + (w + dx)];
                int wi = dy * 11 + dx;
                aq = fmaf(xv, wq[wi], aq);
                ak = fmaf(xv, wk[wi], ak);
                av = fmaf(xv, wv[wi], av);
            }
        }
        long base = ((long)b * NT + c) * HW + p;
        q[base] = aq;
        k[base] = ak;
        vT[((long)b * HW + p) * TP + c] = av;   // transposed, padded ld
    }
}

// ------------- Kernel 2/4: generic wave32 WMMA GEMM  C = A * B^T -------------
// A: M x K (row-major, ldA), B: N x K (row-major, ldB), C: M x N (ldC), batched.
// Block tile 64(M) x 128(N): 8 waves in 2x4, each wave 32x32 (4 accumulators).
#define TM  64
#define TN  128
#define KC  64
#define LDP 68            // LDS row stride: 16B-aligned rows, 4-bank rotation

__global__ __launch_bounds__(256)
void wmma_gemm_rr(const float* __restrict__ A, const float* __restrict__ B,
                  float* __restrict__ C,
                  int M, int N, int K, int ldA, int ldB, int ldC,
                  long sA, long sB, long sC, float scale)
{
    __shared__ __align__(16) float ldsA[2][TM * LDP];   // 2 x 17408 B
    __shared__ __align__(16) float ldsB[2][TN * LDP];   // 2 x 34816 B

    const int tid  = threadIdx.x;
    const int lane = tid & 31;
    const int wave = tid >> 5;
    const int l16  = lane & 15;
    const int half = lane >> 4;
    const int mOff = (wave >> 2) * 32;   // 0 or 32
    const int nOff = (wave & 3) * 32;    // 0,32,64,96

    const int mBase = blockIdx.y * TM;
    const int nBase = blockIdx.x * TN;
    const int b     = blockIdx.z;

    const float* Ab = A + (long)b * sA;
    const float* Bb = B + (long)b * sB;
    float*       Cb = C + (long)b * sC;

    // ---- pre-zero both buffers once: row-OOB regions stay zero for all chunks
    {
        float4 z = make_float4(0.f, 0.f, 0.f, 0.f);
        float4* pa = (float4*)&ldsA[0][0];
        float4* pb = (float4*)&ldsB[0][0];
        for (int i = tid; i < (2 * TM * LDP) / 4; i += 256) pa[i] = z;
        for (int i = tid; i < (2 * TN * LDP) / 4; i += 256) pb[i] = z;
    }
    __syncthreads();

    // ---- staging: async fast path for full-K chunks, sync zero-pad for tail
    auto stage = [&](int cchunk) {
        const int k0 = cchunk * KC;
        float* bufA = ldsA[cchunk & 1];
        float* bufB = ldsB[cchunk & 1];
        if (k0 + KC <= K) {
            // async copies, EXEC-masked for OOB rows (pre-zeroed LDS persists)
#pragma unroll
            for (int i = tid; i < (TM * KC) / 4; i += 256) {
                int r = i >> 4, c4 = (i & 15) << 2;
                int gr = mBase + r;
                if (gr < M) async_copy_b128(Ab + (long)gr * ldA + k0 + c4,
                                            &bufA[r * LDP + c4]);
            }
#pragma unroll
            for (int i = tid; i < (TN * KC) / 4; i += 256) {
                int r = i >> 4, c4 = (i & 15) << 2;
                int gr = nBase + r;
                if (gr < N) async_copy_b128(Bb + (long)gr * ldB + k0 + c4,
                                            &bufB[r * LDP + c4]);
            }
        } else {
            // K-tail chunk: fully overwrite buffer with zero-padded data
            for (int i = tid; i < (TM * KC) / 4; i += 256) {
                int r = i >> 4, c4 = (i & 15) << 2;
                int gr = mBase + r, gk = k0 + c4;
                float4 val = make_float4(0.f, 0.f, 0.f, 0.f);
                if (gr < M) {
                    const float* src = Ab + (long)gr * ldA + gk;  // ld % 4 == 0
                    if (gk + 3 < K) val = *(const float4*)src;
                    else {
                        if (gk + 0 < K) val.x = src[0];
                        if (gk + 1 < K) val.y = src[1];
                        if (gk + 2 < K) val.z = src[2];
                    }
                }
                *(float4*)&bufA[r * LDP + c4] = val;
            }
            for (int i = tid; i < (TN * KC) / 4; i += 256) {
                int r = i >> 4, c4 = (i & 15) << 2;
                int gr = nBase + r, gk = k0 + c4;
                float4 val = make_float4(0.f, 0.f, 0.f, 0.f);
                if (gr < N) {
                    const float* src = Bb + (long)gr * ldB + gk;
                    if (gk + 3 < K) val = *(const float4*)src;
                    else {
                        if (gk + 0 < K) val.x = src[0];
                        if (gk + 1 < K) val.y = src[1];
                        if (gk + 2 < K) val.z = src[2];
                    }
                }
                *(float4*)&bufB[r * LDP + c4] = val;
            }
        }
    };

    v8f acc00 = {0.f,0.f,0.f,0.f,0.f,0.f,0.f,0.f};
    v8f acc01 = acc00, acc10 = acc00, acc11 = acc00;

    const int nc = (K + KC - 1) / KC;
    stage(0);
    async_wait0();
    __syncthreads();

    for (int cchunk = 0; cchunk < nc; ++cchunk) {
        const float* bA = ldsA[cchunk & 1];
        const float* bB = ldsB[cchunk & 1];

        if (cchunk + 1 < nc) stage(cchunk + 1);   // stream next tile during WMMAs

        // ---- 16 steps x 4 chained V_WMMA_F32_16X16X4_F32 per chunk ----
        // A frag: lane L -> row L%16, K = 2*(L/16)+r ; B frag symmetric (N rows).
#pragma unroll
        for (int s = 0; s < KC / 4; ++s) {
            int koff = s * 4 + 2 * half;
            v2f a0 = *(const v2f*)&bA[(mOff      + l16) * LDP + koff];
            v2f a1 = *(const v2f*)&bA[(mOff + 16 + l16) * LDP + koff];
            v2f b0 = *(const v2f*)&bB[(nOff      + l16) * LDP + koff];
            v2f b1 = *(const v2f*)&bB[(nOff + 16 + l16) * LDP + koff];
            acc00 = __builtin_amdgcn_wmma_f32_16x16x4_f32(false, a0, false, b0,
                                                          (short)0, acc00, false, false);
            acc01 = __builtin_amdgcn_wmma_f32_16x16x4_f32(false, a0, false, b1,
                                                          (short)0, acc01, false, false);
            acc10 = __builtin_amdgcn_wmma_f32_16x16x4_f32(false, a1, false, b0,
                                                          (short)0, acc10, false, false);
            acc11 = __builtin_amdgcn_wmma_f32_16x16x4_f32(false, a1, false, b1,
                                                          (short)0, acc11, false, false);
        }
        async_wait0();      // next chunk's async copies have landed in LDS
        __syncthreads();    // all waves done reading current buffer
    }

    // ---- epilogue: C layout VGPR g -> M = g + 8*(lane>=16), N = lane%16 ----
    const int rowB = mBase + mOff + (half ? 8 : 0);
    const int colB = nBase + nOff + l16;
#pragma unroll
    for (int g = 0; g < 8; ++g) {
        int r0 = rowB + g;          // acc00 / acc01
        int r1 = rowB + 16 + g;     // acc10 / acc11
        if (r0 < M) {
            if (colB      < N) Cb[(long)r0 * ldC + colB]      = acc00[g] * scale;
            if (colB + 16 < N) Cb[(long)r0 * ldC + colB + 16] = acc01[g] * scale;
        }
        if (r1 < M) {
            if (colB      < N) Cb[(long)r1 * ldC + colB]      = acc10[g] * scale;
            if (colB + 16 < N) Cb[(long)r1 * ldC + colB + 16] = acc11[g] * scale;
        }
    }
}

// ---------------- Kernel 3: row softmax (one wave32 per row) ----------------
__global__ __launch_bounds__(256)
void softmax_rows(float* __restrict__ alpha, int rows, int n, int ld)
{
    const int row  = blockIdx.x * 8 + (threadIdx.x >> 5);
    const int lane = threadIdx.x & 31;
    if (row >= rows) return;

    float* p = alpha + (long)row * ld;

    float m = -3.402823466e38f;
    for (int j = lane; j < n; j += 32) m = fmaxf(m, p[j]);
#pragma unroll
    for (int o = 16; o > 0; o >>= 1) m = fmaxf(m, __shfl_xor(m, o, 32));

    float s = 0.f;
    for (int j = lane; j < n; j += 32) s += __expf(p[j] - m);
#pragma unroll
    for (int o = 16; o > 0; o >>= 1) s += __shfl_xor(s, o, 32);

    float inv = 1.0f / s;
    for (int j = lane; j < n; j += 32) p[j] = __expf(p[j] - m) * inv;
}

// ---------------------------------------------------------------------------
extern "C" void kernel_launch(void* const* d_in, const int* in_sizes, int n_in,
                              void* d_out, int out_size, void* d_ws, size_t ws_size,
                              hipStream_t stream)
{
    const float* x  = (const float*)d_in[0];
    const float* Wq = (const float*)d_in[1];
    const float* Wk = (const float*)d_in[2];
    const float* Wv = (const float*)d_in[3];
    float* out = (float*)d_out;

    // workspace layout (floats): q | k | vT (padded) | alpha (padded)  ~166 MB
    float* q     = (float*)d_ws;
    float* kbuf  = q    + (size_t)NB * NT * HW;
    float* vT    = kbuf + (size_t)NB * NT * HW;
    float* alpha = vT   + (size_t)NB * HW * TP;

    // 1) depthwise conv -> q, k, vT
    dwconv_qkv<<<dim3(NT, NB), 256, 0, stream>>>(x, Wq, Wk, Wv, q, kbuf, vT);

    // 2) alpha = (q . k^T) / 32     M=N=197, K=1024
    {
        dim3 grid((NT + TN - 1) / TN, (NT + TM - 1) / TM, NB);  // 2 x 4 x 64
        wmma_gemm_rr<<<grid, 256, 0, stream>>>(
            q, kbuf, alpha, NT, NT, HW,
            /*ldA=*/HW, /*ldB=*/HW, /*ldC=*/TP,
            (long)NT * HW, (long)NT * HW, (long)NT * TP, 1.0f / 32.0f);
    }

    // 3) softmax over j for each (b, i)
    softmax_rows<<<(NB * NT + 7) / 8, 256, 0, stream>>>(alpha, NB * NT, NT, TP);

    // 4) out = alpha . v            M=197, N=1024(hw), K=197  (B = vT rows)
    {
        dim3 grid((HW + TN - 1) / TN, (NT + TM - 1) / TM, NB); // 8 x 4 x 64
        wmma_gemm_rr<<<grid, 256, 0, stream>>>(
            alpha, vT, out, NT, HW, NT,
            /*ldA=*/TP, /*ldB=*/TP, /*ldC=*/HW,
            (long)NT * TP, (long)HW * TP, (long)NT * HW, 1.0f);
    }
}